// CosineSimCodebook_1726576854542
// MI455X (gfx1250) — compile-verified
//
#include <hip/hip_runtime.h>
#include <hip/hip_bf16.h>
#include <math.h>

// ---------------------------------------------------------------------------
// CosineSimCodebook for MI455X (gfx1250, wave32, WMMA + async-to-LDS)
//
// Shapes: x[4,2048,512] (bn=8192 rows, d=512), embed[8192,512] (unit rows),
// cluster_size[8192], embed_avg[8192,512].
// Outputs (flat, in order): quantize(4.19M), ind(8192), dist(67.1M),
// new_cluster_size(8192), new_embed_avg(4.19M), new_embed(4.19M).
// ---------------------------------------------------------------------------

typedef __attribute__((ext_vector_type(2))) float v2f;
typedef __attribute__((ext_vector_type(8))) float v8f;
typedef int v4i __attribute__((vector_size(16)));

#define BN    8192          // b*n rows
#define DIM   512           // feature dim (K)
#define CODES 8192          // codebook size
#define DECAY_F 0.8f
#define ONE_MINUS_DECAY 0.2f
#define EPS_F 1e-5f

#if defined(__HIP_DEVICE_COMPILE__) && defined(__gfx1250__) && \
    __has_builtin(__builtin_amdgcn_global_load_async_to_lds_b128)
#define USE_ASYNC_LDS 1
#else
#define USE_ASYNC_LDS 0
#endif

#if USE_ASYNC_LDS
// Async 16B copy global -> LDS, tracked by ASYNCcnt (no VGPR data path).
__device__ __forceinline__ void async_copy_b128(const float* gp, float* lp) {
    __builtin_amdgcn_global_load_async_to_lds_b128(
        (__attribute__((address_space(1))) v4i*)gp,
        (__attribute__((address_space(3))) v4i*)lp,
        0 /*offset imm*/, 0 /*cpol*/);
}
#endif

// ---------------------------------------------------------------------------
// Kernel 1: invnorm[i] = 1 / max(||x_i||, 1e-6)   (one block per row)
// ---------------------------------------------------------------------------
__global__ __launch_bounds__(256) void invnorm_kernel(const float* __restrict__ x,
                                                      float* __restrict__ invn) {
    __shared__ float red[256];
    const int row = blockIdx.x;
    const int tid = threadIdx.x;
    const float* xr = x + (size_t)row * DIM;
    float s = 0.f;
    for (int d = tid; d < DIM; d += 256) { float v = xr[d]; s += v * v; }
    red[tid] = s; __syncthreads();
    for (int off = 128; off > 0; off >>= 1) {
        if (tid < off) red[tid] += red[tid + off];
        __syncthreads();
    }
    if (tid == 0) invn[row] = 1.0f / fmaxf(sqrtf(red[0]), 1e-6f);
}

// ---------------------------------------------------------------------------
// Kernel 2a/2b: EMA init   ncs = DECAY*cluster_size,  nea = DECAY*embed_avg
// ---------------------------------------------------------------------------
__global__ __launch_bounds__(256) void ema_init_cs_kernel(const float* __restrict__ cs,
                                                          float* __restrict__ ncs) {
    int i = blockIdx.x * 256 + threadIdx.x;
    if (i < CODES) ncs[i] = DECAY_F * cs[i];
}
__global__ __launch_bounds__(256) void ema_init_avg_kernel(const float* __restrict__ ea,
                                                           float* __restrict__ nea) {
    size_t i = (size_t)blockIdx.x * 256 + threadIdx.x;
    if (i < (size_t)CODES * DIM) nea[i] = DECAY_F * ea[i];
}

// ---------------------------------------------------------------------------
// Kernel 3: fp32 WMMA GEMM   dist[m][n] = invnorm[m] * sum_k x[m][k]*embed[n][k]
// Block: 256 threads = 8 waves (4 M x 2 N), block tile 128x128,
// wave tile 32x64 = 2x4 accumulators of 16x16, K staged 16-wide in LDS.
// Double-buffered GLOBAL_LOAD_ASYNC_TO_LDS_B128 staging (ASYNCcnt) when
// available; falls back to global->VGPR->ds_store staging otherwise.
// ---------------------------------------------------------------------------
#define TILE_M 128
#define TILE_N 128
#define TILE_K 16

__global__ __launch_bounds__(256) void gemm_dist_kernel(const float* __restrict__ x,
                                                        const float* __restrict__ embed,
                                                        const float* __restrict__ invn,
                                                        float* __restrict__ dist) {
    __shared__ float lA[2][TILE_M * TILE_K];   // x tile, row-major [m][k]
    __shared__ float lB[2][TILE_N * TILE_K];   // embed tile, row-major [n][k]

    const int tid  = threadIdx.x;
    const int lane = tid & 31;
    const int wave = tid >> 5;        // 0..7
    const int wm   = wave & 3;        // wave row   -> m offset wm*32
    const int wn   = wave >> 2;       // wave col   -> n offset wn*64
    const int half = lane >> 4;       // 0 | 1 (K-pair selector per ISA layout)
    const int l16  = lane & 15;

    const int m0 = blockIdx.y * TILE_M;
    const int n0 = blockIdx.x * TILE_N;

    v8f acc[2][4];
#pragma unroll
    for (int mi = 0; mi < 2; ++mi)
#pragma unroll
        for (int ni = 0; ni < 4; ++ni)
#pragma unroll
            for (int r = 0; r < 8; ++r) acc[mi][ni][r] = 0.0f;

    // per-thread tile-load coords: 2048 floats / 256 thr = 8 floats (2x 16B)
    const int ldr = tid >> 1;               // 0..127 tile row
    const int ldc = (tid & 1) * 8;          // 0 or 8  (k within 16)
    const float* gA = &x[(size_t)(m0 + ldr) * DIM + ldc];
    const float* gB = &embed[(size_t)(n0 + ldr) * DIM + ldc];
    const int lofs = ldr * TILE_K + ldc;

#if USE_ASYNC_LDS
    // prime buffer 0 with kt = 0
    async_copy_b128(gA,     &lA[0][lofs]);
    async_copy_b128(gA + 4, &lA[0][lofs + 4]);
    async_copy_b128(gB,     &lB[0][lofs]);
    async_copy_b128(gB + 4, &lB[0][lofs + 4]);
#endif

    int buf = 0;
    for (int kt = 0; kt < DIM; kt += TILE_K) {
#if USE_ASYNC_LDS
        asm volatile("s_wait_asynccnt 0x0" ::: "memory");  // own tile landed in LDS
        __syncthreads();                                   // whole tile visible
        if (kt + TILE_K < DIM) {                           // prefetch next tile
            const int nb = buf ^ 1;
            async_copy_b128(gA + kt + TILE_K,     &lA[nb][lofs]);
            async_copy_b128(gA + kt + TILE_K + 4, &lA[nb][lofs + 4]);
            async_copy_b128(gB + kt + TILE_K,     &lB[nb][lofs]);
            async_copy_b128(gB + kt + TILE_K + 4, &lB[nb][lofs + 4]);
        }
#else
        {   // synchronous staging via VGPRs + ds_store
            const float4* sa = reinterpret_cast<const float4*>(gA + kt);
            float4 a0 = sa[0], a1 = sa[1];
            float4* da = reinterpret_cast<float4*>(&lA[0][lofs]);
            da[0] = a0; da[1] = a1;
            const float4* sb = reinterpret_cast<const float4*>(gB + kt);
            float4 b0 = sb[0], b1 = sb[1];
            float4* db = reinterpret_cast<float4*>(&lB[0][lofs]);
            db[0] = b0; db[1] = b1;
        }
        __syncthreads();
#endif

        const float* tileA = lA[buf];
        const float* tileB = lB[buf];
#pragma unroll
        for (int kb = 0; kb < TILE_K; kb += 4) {
#pragma unroll
            for (int ni = 0; ni < 4; ++ni) {
                const int nrow = wn * 64 + ni * 16 + l16;
                v2f bf;
                bf.x = tileB[nrow * TILE_K + kb + 2 * half];
                bf.y = tileB[nrow * TILE_K + kb + 2 * half + 1];
#pragma unroll
                for (int mi = 0; mi < 2; ++mi) {
                    const int mrow = wm * 32 + mi * 16 + l16;
                    v2f af;
                    af.x = tileA[mrow * TILE_K + kb + 2 * half];
                    af.y = tileA[mrow * TILE_K + kb + 2 * half + 1];
                    acc[mi][ni] = __builtin_amdgcn_wmma_f32_16x16x4_f32(
                        false, af, false, bf, (short)0, acc[mi][ni], false, false);
                }
            }
        }
#if USE_ASYNC_LDS
        buf ^= 1;
#else
        __syncthreads();      // single-buffer path: protect LDS before restage
#endif
    }

    // store D: VGPR r holds M = r (lanes 0-15) / r+8 (lanes 16-31), N = l16
#pragma unroll
    for (int mi = 0; mi < 2; ++mi) {
#pragma unroll
        for (int r = 0; r < 8; ++r) {
            const int grow = m0 + wm * 32 + mi * 16 + r + 8 * half;
            const float scale = invn[grow];
#pragma unroll
            for (int ni = 0; ni < 4; ++ni) {
                const int gcol = n0 + wn * 64 + ni * 16 + l16;
                dist[(size_t)grow * CODES + gcol] = acc[mi][ni][r] * scale;
            }
        }
    }
}

// ---------------------------------------------------------------------------
// Kernel 4: per-row argmax over 8192 logits -> ind, quantize = embed[ind],
// and EMA scatter: ncs[ind] += 0.2, nea[ind][:] += 0.2 * xnorm_row
// ---------------------------------------------------------------------------
__global__ __launch_bounds__(256) void argmax_quant_kernel(const float* __restrict__ dist,
                                                           const float* __restrict__ x,
                                                           const float* __restrict__ invn,
                                                           const float* __restrict__ embed,
                                                           float* __restrict__ quant,
                                                           float* __restrict__ indf,
                                                           float* __restrict__ ncs,
                                                           float* __restrict__ nea) {
    __shared__ float sv[256];
    __shared__ int   si[256];
    const int row = blockIdx.x;
    const int tid = threadIdx.x;
    const float* drow = dist + (size_t)row * CODES;

    float best = -INFINITY; int bidx = 0x7fffffff;
    for (int c = tid; c < CODES; c += 256) {
        float v = drow[c];
        if (v > best) { best = v; bidx = c; }   // stride scan: c increasing
    }
    sv[tid] = best; si[tid] = bidx; __syncthreads();
    for (int off = 128; off > 0; off >>= 1) {
        if (tid < off) {
            float ov = sv[tid + off]; int oi = si[tid + off];
            if (ov > sv[tid] || (ov == sv[tid] && oi < si[tid])) { sv[tid] = ov; si[tid] = oi; }
        }
        __syncthreads();
    }
    const int bi = si[0];
    if (tid == 0) {
        indf[row] = (float)bi;                       // argmax index
        atomicAdd(&ncs[bi], ONE_MINUS_DECAY);        // bins EMA contribution
    }
    const float inv = invn[row];
    const float* erow = embed + (size_t)bi * DIM;
    const float* xrow = x + (size_t)row * DIM;
    float* qrow = quant + (size_t)row * DIM;
    float* arow = nea + (size_t)bi * DIM;
    for (int d = tid; d < DIM; d += 256) {
        qrow[d] = erow[d];                                       // gather quantize
        atomicAdd(&arow[d], ONE_MINUS_DECAY * xrow[d] * inv);    // embed_sum EMA
    }
}

// ---------------------------------------------------------------------------
// Kernel 5: denom = sum(new_cluster_size)   (single block)
// ---------------------------------------------------------------------------
__global__ __launch_bounds__(256) void denom_kernel(const float* __restrict__ ncs,
                                                    float* __restrict__ denom) {
    __shared__ float red[256];
    const int tid = threadIdx.x;
    float s = 0.f;
    for (int c = tid; c < CODES; c += 256) s += ncs[c];
    red[tid] = s; __syncthreads();
    for (int off = 128; off > 0; off >>= 1) {
        if (tid < off) red[tid] += red[tid + off];
        __syncthreads();
    }
    if (tid == 0) denom[0] = red[0];
}

// ---------------------------------------------------------------------------
// Kernel 6: smoothed = (ncs+eps)/(denom+C*eps)*denom;
//           new_embed = l2norm(nea / smoothed)       (one block per code)
// ---------------------------------------------------------------------------
__global__ __launch_bounds__(256) void new_embed_kernel(const float* __restrict__ ncs,
                                                        const float* __restrict__ nea,
                                                        const float* __restrict__ denom,
                                                        float* __restrict__ nemb) {
    __shared__ float red[256];
    const int c   = blockIdx.x;
    const int tid = threadIdx.x;
    const float dn = denom[0];
    const float smoothed = (ncs[c] + EPS_F) / (dn + (float)CODES * EPS_F) * dn;
    const float invs = 1.0f / smoothed;
    const float* arow = nea + (size_t)c * DIM;
    float ss = 0.f;
    for (int d = tid; d < DIM; d += 256) { float t = arow[d] * invs; ss += t * t; }
    red[tid] = ss; __syncthreads();
    for (int off = 128; off > 0; off >>= 1) {
        if (tid < off) red[tid] += red[tid + off];
        __syncthreads();
    }
    const float scale = invs / fmaxf(sqrtf(red[0]), 1e-6f);
    float* orow = nemb + (size_t)c * DIM;
    for (int d = tid; d < DIM; d += 256) orow[d] = arow[d] * scale;
}

// ---------------------------------------------------------------------------
extern "C" void kernel_launch(void* const* d_in, const int* in_sizes, int n_in,
                              void* d_out, int out_size, void* d_ws, size_t ws_size,
                              hipStream_t stream) {
    const float* x     = (const float*)d_in[0];   // [4,2048,512]
    const float* embed = (const float*)d_in[1];   // [1,8192,512]
    const float* cs    = (const float*)d_in[2];   // [1,8192]
    const float* ea    = (const float*)d_in[3];   // [1,8192,512]

    float* out = (float*)d_out;
    float* quant = out;                                   // 4,194,304
    float* indf  = out + (size_t)BN * DIM;                // 8,192
    float* dist  = indf + BN;                             // 67,108,864
    float* ncs   = dist + (size_t)BN * CODES;             // 8,192
    float* nea   = ncs + CODES;                           // 4,194,304
    float* nemb  = nea + (size_t)CODES * DIM;             // 4,194,304

    float* invn  = (float*)d_ws;        // 8192 floats
    float* denom = invn + BN;           // 1 float

    // 1) row inverse norms of x
    invnorm_kernel<<<BN, 256, 0, stream>>>(x, invn);
    // 2) EMA init (DECAY * old) — atomics in kernel 4 accumulate on top
    ema_init_cs_kernel<<<(CODES + 255) / 256, 256, 0, stream>>>(cs, ncs);
    ema_init_avg_kernel<<<((size_t)CODES * DIM + 255) / 256, 256, 0, stream>>>(ea, nea);
    // 3) fp32 WMMA GEMM: dist = (x @ embed^T) * invnorm[row]
    gemm_dist_kernel<<<dim3(CODES / TILE_N, BN / TILE_M), 256, 0, stream>>>(x, embed, invn, dist);
    // 4) argmax + gather quantize + EMA scatter
    argmax_quant_kernel<<<BN, 256, 0, stream>>>(dist, x, invn, embed, quant, indf, ncs, nea);
    // 5) denom reduction
    denom_kernel<<<1, 256, 0, stream>>>(ncs, denom);
    // 6) laplace smoothing + l2norm -> new_embed
    new_embed_kernel<<<CODES, 256, 0, stream>>>(ncs, nea, denom, nemb);
}